// SentenceAwareSelfAttention_14053132992614
// MI455X (gfx1250) — compile-verified
//
#include <hip/hip_runtime.h>
#include <hip/hip_bf16.h>

// B=8, N=64, S=128, D=512, H=8, DK=64
#define Bv 8
#define Nv 64
#define Sv 128
#define Dv 512
#define Hv 8
#define DKv 64
#define RN (Bv * Nv)            // 512 sentence rows
#define INV_SQRT_DK 0.125f

typedef float v2f __attribute__((ext_vector_type(2)));
typedef float v8f __attribute__((ext_vector_type(8)));

#if __has_builtin(__builtin_amdgcn_global_load_async_to_lds_b128)
#define USE_ASYNC_COPY 1
typedef int v4i_vs __attribute__((vector_size(16)));
typedef v4i_vs __attribute__((address_space(1))) gv4i;   // global (__device__) int4
typedef v4i_vs __attribute__((address_space(3))) lv4i;   // LDS (__shared__) int4
#endif

// -------------------------------------------------------------------------
// Kernel 0: fold sent_q into Wk:  wk_eff[h][d] = sum_k Wk[d, h*64+k]*sent_q[h,k]
//           bk_dot[h] = sum_k bk[h*64+k]*sent_q[h,k]
// -------------------------------------------------------------------------
__global__ void k_fold(const float* __restrict__ Wk, const float* __restrict__ bk,
                       const float* __restrict__ sq,
                       float* __restrict__ wk_eff, float* __restrict__ bk_dot) {
    int t = blockIdx.x * 256 + threadIdx.x;   // 4096 threads: (h,d)
    int h = t >> 9;
    int d = t & 511;
    float acc = 0.f;
    #pragma unroll 8
    for (int k = 0; k < DKv; ++k)
        acc += Wk[d * Dv + h * DKv + k] * sq[h * DKv + k];
    wk_eff[h * Dv + d] = acc;                 // layout [h][d]
    if (t < Hv) {
        float a2 = 0.f;
        for (int k = 0; k < DKv; ++k) a2 += bk[t * DKv + k] * sq[t * DKv + k];
        bk_dot[t] = a2;
    }
}

// -------------------------------------------------------------------------
// Kernel A: one workgroup per sentence (b,n). Stage x tile (128x512 fp32,
// 256 KiB) in LDS, compute masked sentence scores, softmax per head (one
// wave per head, wave32 shuffles), attention-pool x, project by Wv head
// block + bv -> rep[r, 0:512].
// -------------------------------------------------------------------------
__global__ void __launch_bounds__(256)
k_sent(const float* __restrict__ x, const int* __restrict__ mask,
       const float* __restrict__ wk_eff, const float* __restrict__ bk_dot,
       const float* __restrict__ Wv, const float* __restrict__ bv,
       float* __restrict__ rep) {
    extern __shared__ float xs[];             // Sv*Dv = 65536 floats (256 KiB)
    __shared__ float wkl[Hv * Dv];            // 16 KiB
    __shared__ float sc[Hv * Sv];             // 4 KiB  (scores -> attn)
    __shared__ float pooled[Hv * Dv];         // 16 KiB

    const int r    = blockIdx.x;              // b*N + n
    const int tid  = threadIdx.x;
    const int wave = tid >> 5;
    const int lane = tid & 31;
    const float* xg = x + (size_t)r * Sv * Dv;

    // stage folded key weights
    for (int i = tid; i < Hv * Dv; i += 256) wkl[i] = wk_eff[i];

    // stage x tile into LDS (async DMA path if the toolchain exposes it)
#ifdef USE_ASYNC_COPY
    {
        gv4i* g = (gv4i*)xg;
        lv4i* l = (lv4i*)xs;
        for (int i = tid; i < (Sv * Dv) / 4; i += 256)
            __builtin_amdgcn_global_load_async_to_lds_b128(g + i, l + i, 0, 0);
#if __has_builtin(__builtin_amdgcn_s_wait_asynccnt)
        __builtin_amdgcn_s_wait_asynccnt(0);
#else
        asm volatile("s_wait_asynccnt 0" ::: "memory");
#endif
    }
#else
    for (int i = tid; i < (Sv * Dv) / 4; i += 256)
        ((float4*)xs)[i] = ((const float4*)xg)[i];
#endif
    __syncthreads();

    // ---- scores: wave w owns head h=w; cooperative 512-dot per token ----
    {
        const int h = wave;
        const float bkd = bk_dot[h];
        for (int s = 0; s < Sv; ++s) {
            float p = 0.f;
            #pragma unroll 4
            for (int d = lane; d < Dv; d += 32)
                p += xs[s * Dv + d] * wkl[h * Dv + d];
            for (int off = 16; off; off >>= 1) p += __shfl_xor(p, off, 32);
            if (lane == 0) {
                float scv = (p + bkd) * INV_SQRT_DK;
                if (mask[r * Sv + s] == 0) scv = -1e9f;
                sc[h * Sv + s] = scv;
            }
        }
    }
    __syncthreads();

    // ---- softmax over S=128 per head (wave w == head w) ----
    {
        const int h = wave;
        float v0 = sc[h * Sv + lane +  0];
        float v1 = sc[h * Sv + lane + 32];
        float v2 = sc[h * Sv + lane + 64];
        float v3 = sc[h * Sv + lane + 96];
        float m = fmaxf(fmaxf(v0, v1), fmaxf(v2, v3));
        for (int off = 16; off; off >>= 1) m = fmaxf(m, __shfl_xor(m, off, 32));
        float e0 = __expf(v0 - m), e1 = __expf(v1 - m);
        float e2 = __expf(v2 - m), e3 = __expf(v3 - m);
        float sum = e0 + e1 + e2 + e3;
        for (int off = 16; off; off >>= 1) sum += __shfl_xor(sum, off, 32);
        float inv = 1.f / sum;
        sc[h * Sv + lane +  0] = e0 * inv;
        sc[h * Sv + lane + 32] = e1 * inv;
        sc[h * Sv + lane + 64] = e2 * inv;
        sc[h * Sv + lane + 96] = e3 * inv;
    }
    __syncthreads();

    // ---- attention pooling of x (conflict-free: lanes on adjacent d) ----
    {
        const int d0 = tid * 2;
        for (int h = 0; h < Hv; ++h) {
            float a0 = 0.f, a1 = 0.f;
            for (int s = 0; s < Sv; ++s) {
                float a = sc[h * Sv + s];
                a0 += a * xs[s * Dv + d0];
                a1 += a * xs[s * Dv + d0 + 1];
            }
            pooled[h * Dv + d0]     = a0;
            pooled[h * Dv + d0 + 1] = a1;
        }
    }
    __syncthreads();

    // ---- rep[r, h*64+dk] = pooled[h] . Wv[:, h*64+dk] + bv ----
    {
        const int col0 = tid * 2;             // wave w covers head w's 64 cols
        const int h = col0 >> 6;
        float a0 = 0.f, a1 = 0.f;
        for (int d = 0; d < Dv; ++d) {
            float p = pooled[h * Dv + d];     // broadcast within wave
            a0 += p * Wv[d * Dv + col0];
            a1 += p * Wv[d * Dv + col0 + 1];
        }
        rep[(size_t)r * Dv + col0]     = a0 + bv[col0];
        rep[(size_t)r * Dv + col0 + 1] = a1 + bv[col0 + 1];
    }
}

// -------------------------------------------------------------------------
// Kernel C: O = rep (512x512) @ Wo (512x512) + bo  via V_WMMA_F32_16X16X4_F32
// One wave per 16x16 tile; 8 waves/block; 128 blocks for 32x32 tiles.
// A 16x4 frag: lanes0-15 hold M=lane K={k,k+1}; lanes16-31 K={k+2,k+3}.
// B 4x16 frag: VGPR0 = rows K(+hi*2), VGPR1 = rows K(+hi*2+1), N=lane%16.
// C/D 16x16:  VGPR i -> M = hi*8 + i, N = lane%16.
// -------------------------------------------------------------------------
__global__ void __launch_bounds__(256)
k_wo(const float* __restrict__ rep, const float* __restrict__ Wo,
     const float* __restrict__ bo, float* __restrict__ O) {
    const int tid  = threadIdx.x;
    const int wave = tid >> 5;
    const int lane = tid & 31;
    const int tile = blockIdx.x * 8 + wave;   // 1024 tiles
    const int tm = tile >> 5;
    const int tn = tile & 31;
    const int lo = lane & 15;
    const int hi = lane >> 4;
    const int arow = tm * 16 + lo;
    const int bcol = tn * 16 + lo;

    v8f c = {};
#if __has_builtin(__builtin_amdgcn_wmma_f32_16x16x4_f32)
    for (int k = 0; k < Dv; k += 16) {
        #pragma unroll
        for (int kk = 0; kk < 16; kk += 4) {
            const int k0 = k + kk + hi * 2;
            v2f a = *(const v2f*)(rep + (size_t)arow * Dv + k0);
            v2f b;
            b.x = Wo[(size_t)(k0)     * Dv + bcol];
            b.y = Wo[(size_t)(k0 + 1) * Dv + bcol];
            c = __builtin_amdgcn_wmma_f32_16x16x4_f32(
                    /*neg_a=*/false, a, /*neg_b=*/false, b,
                    /*c_mod=*/(short)0, c, /*reuse_a=*/false, /*reuse_b=*/false);
        }
    }
#else
    #pragma unroll
    for (int i = 0; i < 8; ++i) {
        const int row = tm * 16 + hi * 8 + i;
        float acc = 0.f;
        for (int k = 0; k < Dv; ++k)
            acc += rep[(size_t)row * Dv + k] * Wo[(size_t)k * Dv + bcol];
        c[i] = acc;
    }
#endif
    const float bias = bo[bcol];
    #pragma unroll
    for (int i = 0; i < 8; ++i) {
        const int row = tm * 16 + hi * 8 + i;
        O[(size_t)row * Dv + bcol] = c[i] + bias;
    }
}

// -------------------------------------------------------------------------
// Kernel D: broadcast each O row to all S=128 tokens (coalesced float4).
// out[(r*S+s)*D + d] = O[r*D + d]. 32768 blocks x 256 threads, 2 rows/block.
// -------------------------------------------------------------------------
__global__ void __launch_bounds__(256)
k_bcast(const float4* __restrict__ O4, float4* __restrict__ out4) {
    const size_t rs  = (size_t)blockIdx.x * 2 + (threadIdx.x >> 7); // r*128+s
    const int    c   = threadIdx.x & 127;                           // float4 col
    const size_t r   = rs >> 7;
    out4[rs * (Dv / 4) + c] = O4[r * (Dv / 4) + c];
}

extern "C" void kernel_launch(void* const* d_in, const int* in_sizes, int n_in,
                              void* d_out, int out_size, void* d_ws, size_t ws_size,
                              hipStream_t stream) {
    (void)in_sizes; (void)n_in; (void)out_size; (void)ws_size;
    const float* x    = (const float*)d_in[0];
    const int*   mask = (const int*)  d_in[1];
    // d_in[2] = Wq, d_in[3] = bq: provably dead (token softmax over singleton)
    const float* Wk   = (const float*)d_in[4];
    const float* bk   = (const float*)d_in[5];
    const float* Wv   = (const float*)d_in[6];
    const float* bv   = (const float*)d_in[7];
    const float* sq   = (const float*)d_in[8];
    const float* Wo   = (const float*)d_in[9];
    const float* bo   = (const float*)d_in[10];
    float* out = (float*)d_out;

    float* ws     = (float*)d_ws;
    float* wk_eff = ws;                      // 4096 floats
    float* bk_dot = ws + 4096;               // 8 (padded to 16)
    float* rep    = ws + 4112;               // 512*512
    float* O      = rep + RN * Dv;           // 512*512

    k_fold<<<16, 256, 0, stream>>>(Wk, bk, sq, wk_eff, bk_dot);
    k_sent<<<RN, 256, Sv * Dv * sizeof(float), stream>>>(
        x, mask, wk_eff, bk_dot, Wv, bv, rep);
    k_wo<<<128, 256, 0, stream>>>(rep, Wo, bo, O);
    k_bcast<<<(RN * Sv) / 2, 256, 0, stream>>>((const float4*)O, (float4*)out);
}